// twist_66322884985439
// MI455X (gfx1250) — compile-verified
//
#include <hip/hip_runtime.h>
#include <math.h>

// fp32, memory-bound SE(3) chain kernel for MI455X (gfx1250, wave32).
// Stage 1: tiny setup kernel folds all joint-invariant math (norm, skew,
//          W^2, W v̂, W² v̂, and the θ>1e-9 select) into 2x25 scalars in d_ws.
// Stage 2: one thread per element; constants come in via uniform s_load
//          (SGPR-resident), data via 64/128-bit non-temporal global ops
//          (streamed-once working set ~360MB > 192MB L2). Branch-free
//          FMA Cody-Waite sincos (|phi| <= ~10 here; reduction valid far
//          beyond) replaces the heavyweight OCML sincos.

typedef float v4f __attribute__((ext_vector_type(4)));
typedef float v2f __attribute__((ext_vector_type(2)));

// Joint constant layout in d_ws (floats)
#define JW0 0    // w[3]
#define JV0 3    // v[3]
#define JVN 6    // v̂[3]        (= v if !rot)
#define JTH 9    // theta       (= 1 if !rot)
#define JWN 10   // ŵ[3]        (= 0 if !rot)
#define JB 13    // W² sym: B00,B01,B02,B11,B12,B22  (= 0 if !rot)
#define JA1 19   // W  v̂ [3]    (= 0 if !rot)
#define JA2 22   // W² v̂ [3]    (= 0 if !rot)
#define JSTRIDE 32

__global__ void joint_setup_kernel(const float* __restrict__ xiRev,
                                   const float* __restrict__ xiPri,
                                   float* __restrict__ ws) {
  const int j = threadIdx.x;
  if (j >= 2) return;
  const float* xi = (j == 0) ? xiRev : xiPri;
  float* o = ws + j * JSTRIDE;

  const float w0 = xi[0], w1 = xi[1], w2 = xi[2];
  const float v0 = xi[3], v1 = xi[4], v2 = xi[5];
  o[JW0 + 0] = w0; o[JW0 + 1] = w1; o[JW0 + 2] = w2;
  o[JV0 + 0] = v0; o[JV0 + 1] = v1; o[JV0 + 2] = v2;

  const float th  = sqrtf(w0 * w0 + w1 * w1 + w2 * w2);
  const float inv = 1.0f / fmaxf(th, 1e-12f);
  const float a = w0 * inv, b = w1 * inv, c = w2 * inv;
  const float vn0 = v0 * inv, vn1 = v1 * inv, vn2 = v2 * inv;

  // W = skew(ŵ), W2 = W @ W (explicit fp32 matmul, matches reference; symmetric)
  const float W[3][3] = {{0.0f, -c, b}, {c, 0.0f, -a}, {-b, a, 0.0f}};
  float B[3][3];
  for (int i = 0; i < 3; ++i)
    for (int k = 0; k < 3; ++k)
      B[i][k] = W[i][0] * W[0][k] + W[i][1] * W[1][k] + W[i][2] * W[2][k];

  const float a10 = W[0][0]*vn0 + W[0][1]*vn1 + W[0][2]*vn2;
  const float a11 = W[1][0]*vn0 + W[1][1]*vn1 + W[1][2]*vn2;
  const float a12 = W[2][0]*vn0 + W[2][1]*vn1 + W[2][2]*vn2;
  const float a20 = B[0][0]*vn0 + B[0][1]*vn1 + B[0][2]*vn2;
  const float a21 = B[1][0]*vn0 + B[1][1]*vn1 + B[1][2]*vn2;
  const float a22 = B[2][0]*vn0 + B[2][1]*vn1 + B[2][2]*vn2;

  const bool rot = th > 1e-9f;
  if (rot) {
    o[JTH] = th;
    o[JWN + 0] = a;   o[JWN + 1] = b;   o[JWN + 2] = c;
    o[JVN + 0] = vn0; o[JVN + 1] = vn1; o[JVN + 2] = vn2;
    o[JB + 0] = B[0][0]; o[JB + 1] = B[0][1]; o[JB + 2] = B[0][2];
    o[JB + 3] = B[1][1]; o[JB + 4] = B[1][2]; o[JB + 5] = B[2][2];
    o[JA1 + 0] = a10; o[JA1 + 1] = a11; o[JA1 + 2] = a12;
    o[JA2 + 0] = a20; o[JA2 + 1] = a21; o[JA2 + 2] = a22;
  } else {
    // Fallback fold: phi = q, R_rod = I, p_rod = phi * v  ==  reference's
    // (R=I, p=q*v) branch, with zero per-element overhead in the main kernel.
    o[JTH] = 1.0f;
    o[JWN + 0] = 0.0f; o[JWN + 1] = 0.0f; o[JWN + 2] = 0.0f;
    o[JVN + 0] = v0;   o[JVN + 1] = v1;   o[JVN + 2] = v2;
    for (int i = 0; i < 6; ++i) o[JB + i] = 0.0f;
    for (int i = 0; i < 3; ++i) { o[JA1 + i] = 0.0f; o[JA2 + i] = 0.0f; }
  }
}

namespace {

// Branch-free sincos: 3-term FMA Cody-Waite reduction (accurate to |x| in the
// thousands; here |x| <= ~10) + Cephes minimax polynomials on |t| <= pi/4,
// quadrant fixup via two's-complement sign injection. ~1-2 ulp.
__device__ __forceinline__ void fast_sincosf(float x, float* sp, float* cp) {
  const float j = __builtin_rintf(x * 0.63661977236758134f);  // x * 2/pi
  float t = fmaf(j, -1.5707962512969971e+00f, x);
  t = fmaf(j, -7.5497894158615964e-08f, t);
  t = fmaf(j, -5.3903029534742384e-15f, t);
  const int q = (int)j;
  const float t2 = t * t;

  float ps = fmaf(-1.9515295891e-04f, t2, 8.3321608736e-03f);
  ps = fmaf(ps, t2, -1.6666654611e-01f);
  ps = fmaf(ps * t2, t, t);                       // sin(t)

  float pc = fmaf(2.4433157e-05f, t2, -1.3887316e-03f);
  pc = fmaf(pc, t2, 4.1666645e-02f);
  pc = fmaf(pc, t2, -5.0e-01f);
  pc = fmaf(pc, t2, 1.0f);                        // cos(t)

  const bool swap = (q & 1) != 0;
  const float s0 = swap ? pc : ps;
  const float c0 = swap ? ps : pc;
  const unsigned ssign = ((unsigned)q << 30) & 0x80000000u;        // bit1(q)
  const unsigned csign = ((unsigned)(q + 1) << 30) & 0x80000000u;  // bit1(q+1)
  *sp = __uint_as_float(__float_as_uint(s0) ^ ssign);
  *cp = __uint_as_float(__float_as_uint(c0) ^ csign);
}

// twist = [R^T w ; R^T (v - p x w)]
__device__ __forceinline__ void body_twist_dev(const float R[3][3], const float p[3],
                                               const float* __restrict__ J, float out[6]) {
  const float w0 = J[JW0], w1 = J[JW0 + 1], w2 = J[JW0 + 2];
  const float v0 = J[JV0], v1 = J[JV0 + 1], v2 = J[JV0 + 2];
#pragma unroll
  for (int i = 0; i < 3; ++i)
    out[i] = fmaf(R[2][i], w2, fmaf(R[1][i], w1, R[0][i] * w0));
  const float u0 = fmaf(p[2], w1, fmaf(-p[1], w2, v0));
  const float u1 = fmaf(p[0], w2, fmaf(-p[2], w0, v1));
  const float u2 = fmaf(p[1], w0, fmaf(-p[0], w1, v2));
#pragma unroll
  for (int i = 0; i < 3; ++i)
    out[3 + i] = fmaf(R[2][i], u2, fmaf(R[1][i], u1, R[0][i] * u0));
}

// T <- T @ [[I + sW + c1*W², phi*v̂ + c1*(W v̂) + (phi-s)*(W² v̂)],[0 0 0 1]]
// (skew/symmetric structure expanded; rot-select pre-folded into constants)
__device__ __forceinline__ void apply_rod(float R[3][3], float p[3], float qv,
                                          const float* __restrict__ J) {
  const float phi = qv * J[JTH];
  float s, cv;
  fast_sincosf(phi, &s, &cv);
  const float c1 = 1.0f - cv;
  const float ps = phi - s;

  const float sx = s * J[JWN + 0];
  const float sy = s * J[JWN + 1];
  const float sz = s * J[JWN + 2];
  const float B00 = J[JB + 0], B01 = J[JB + 1], B02 = J[JB + 2];
  const float B11 = J[JB + 3], B12 = J[JB + 4], B22 = J[JB + 5];

  float Rr[3][3];
  Rr[0][0] = fmaf(c1, B00, 1.0f);
  Rr[0][1] = fmaf(c1, B01, -sz);
  Rr[0][2] = fmaf(c1, B02,  sy);
  Rr[1][0] = fmaf(c1, B01,  sz);
  Rr[1][1] = fmaf(c1, B11, 1.0f);
  Rr[1][2] = fmaf(c1, B12, -sx);
  Rr[2][0] = fmaf(c1, B02, -sy);
  Rr[2][1] = fmaf(c1, B12,  sx);
  Rr[2][2] = fmaf(c1, B22, 1.0f);

  float pr[3];
#pragma unroll
  for (int i = 0; i < 3; ++i)
    pr[i] = fmaf(phi, J[JVN + i], fmaf(c1, J[JA1 + i], ps * J[JA2 + i]));

  float Rn[3][3], pn[3];
#pragma unroll
  for (int i = 0; i < 3; ++i) {
#pragma unroll
    for (int j = 0; j < 3; ++j)
      Rn[i][j] = fmaf(R[i][2], Rr[2][j], fmaf(R[i][1], Rr[1][j], R[i][0] * Rr[0][j]));
    pn[i] = fmaf(R[i][2], pr[2], fmaf(R[i][1], pr[1], fmaf(R[i][0], pr[0], p[i])));
  }
#pragma unroll
  for (int i = 0; i < 3; ++i) {
#pragma unroll
    for (int j = 0; j < 3; ++j) R[i][j] = Rn[i][j];
    p[i] = pn[i];
  }
}

} // namespace

__global__ __launch_bounds__(256) void se3_chain_kernel(
    const float* __restrict__ Tin, const float* __restrict__ Q,
    const float* __restrict__ JC,
    float* __restrict__ outRev, float* __restrict__ outPri,
    float* __restrict__ outT, int N)
{
  const int n = blockIdx.x * blockDim.x + threadIdx.x;
  if (n >= N) return;

  const float* JR = JC;             // uniform -> s_load -> SGPRs
  const float* JP = JC + JSTRIDE;

  // ---- load T (rows 0..2; row 3 constant) and q, non-temporal ----
  const v4f* tv = reinterpret_cast<const v4f*>(Tin) + 4 * (size_t)n;
  const v4f r0 = __builtin_nontemporal_load(tv + 0);
  const v4f r1 = __builtin_nontemporal_load(tv + 1);
  const v4f r2 = __builtin_nontemporal_load(tv + 2);
  const v2f qq = __builtin_nontemporal_load(reinterpret_cast<const v2f*>(Q) + (size_t)n);

  float R[3][3] = {{r0.x, r0.y, r0.z}, {r1.x, r1.y, r1.z}, {r2.x, r2.y, r2.z}};
  float p[3]    = {r0.w, r1.w, r2.w};

  float tw[6];

  // ---- revolute twist (input T), then T <- T @ rod(rev, q0) ----
  body_twist_dev(R, p, JR, tw);
  {
    v2f* o = reinterpret_cast<v2f*>(outRev) + 3 * (size_t)n;
    v2f o0 = {tw[0], tw[1]}, o1 = {tw[2], tw[3]}, o2 = {tw[4], tw[5]};
    __builtin_nontemporal_store(o0, o + 0);
    __builtin_nontemporal_store(o1, o + 1);
    __builtin_nontemporal_store(o2, o + 2);
  }
  apply_rod(R, p, qq.x, JR);

  // ---- prismatic twist (updated T), then T <- T @ rod(pri, q1) ----
  body_twist_dev(R, p, JP, tw);
  {
    v2f* o = reinterpret_cast<v2f*>(outPri) + 3 * (size_t)n;
    v2f o0 = {tw[0], tw[1]}, o1 = {tw[2], tw[3]}, o2 = {tw[4], tw[5]};
    __builtin_nontemporal_store(o0, o + 0);
    __builtin_nontemporal_store(o1, o + 1);
    __builtin_nontemporal_store(o2, o + 2);
  }
  apply_rod(R, p, qq.y, JP);

  // ---- store final T (4 x b128, non-temporal) ----
  {
    v4f* o = reinterpret_cast<v4f*>(outT) + 4 * (size_t)n;
    v4f o0 = {R[0][0], R[0][1], R[0][2], p[0]};
    v4f o1 = {R[1][0], R[1][1], R[1][2], p[1]};
    v4f o2 = {R[2][0], R[2][1], R[2][2], p[2]};
    v4f o3 = {0.0f, 0.0f, 0.0f, 1.0f};
    __builtin_nontemporal_store(o0, o + 0);
    __builtin_nontemporal_store(o1, o + 1);
    __builtin_nontemporal_store(o2, o + 2);
    __builtin_nontemporal_store(o3, o + 3);
  }
}

extern "C" void kernel_launch(void* const* d_in, const int* in_sizes, int n_in,
                              void* d_out, int out_size, void* d_ws, size_t ws_size,
                              hipStream_t stream) {
  const float* T   = (const float*)d_in[0];  // (N,4,4)
  const float* Q   = (const float*)d_in[1];  // (N,2)
  const float* rev = (const float*)d_in[2];  // (6,)
  const float* pri = (const float*)d_in[3];  // (6,)
  const int N = in_sizes[0] / 16;

  float* out  = (float*)d_out;
  float* oRev = out;                         // (N,6)
  float* oPri = out + (size_t)6 * N;         // (N,6)
  float* oT   = out + (size_t)12 * N;        // (N,4,4)
  float* jc   = (float*)d_ws;                // 2 * JSTRIDE floats

  joint_setup_kernel<<<1, 32, 0, stream>>>(rev, pri, jc);

  const int block = 256;
  const int grid  = (N + block - 1) / block;
  se3_chain_kernel<<<grid, block, 0, stream>>>(T, Q, jc, oRev, oPri, oT, N);
}